// TripletLoss_78426102825681
// MI455X (gfx1250) — compile-verified
//
#include <hip/hip_runtime.h>

#define B_ROWS 4096
#define D_DIM  512
#define MARGIN 0.2f
#define NT 8   // 16-wide col tiles per wave -> 128 cols per wave

typedef __attribute__((ext_vector_type(2))) float v2f;
typedef __attribute__((ext_vector_type(8))) float v8f;

// All distance values are >= 0 (and +inf sentinel), so non-negative IEEE floats
// order identically to their unsigned-int bit patterns.
__device__ __forceinline__ void atomic_min_f32_nonneg(float* addr, float v) {
    atomicMin((unsigned int*)addr, __float_as_uint(v));
}

// ---------------------------------------------------------------------------
// Kernel 1: per-row squared norms, d_ap (squared, as in the reference), and
// init of the min-mining buffers. One wave per row.
// ---------------------------------------------------------------------------
__global__ void __launch_bounds__(256) prep_kernel(
    const float* __restrict__ anchor, const float* __restrict__ positive,
    float* __restrict__ sq, float* __restrict__ dap,
    float* __restrict__ minL, float* __restrict__ fback)
{
    const int wave = threadIdx.x >> 5;
    const int lane = threadIdx.x & 31;
    const int row  = blockIdx.x * 8 + wave;

    const float4* a4 = (const float4*)(anchor   + (size_t)row * D_DIM);
    const float4* p4 = (const float4*)(positive + (size_t)row * D_DIM);

    float ssq = 0.0f, sd = 0.0f;
#pragma unroll
    for (int e = 0; e < 4; ++e) {
        float4 a = a4[e * 32 + lane];
        float4 p = p4[e * 32 + lane];
        ssq += a.x*a.x + a.y*a.y + a.z*a.z + a.w*a.w;
        float dx = a.x-p.x, dy = a.y-p.y, dz = a.z-p.z, dw = a.w-p.w;
        sd  += dx*dx + dy*dy + dz*dz + dw*dw;
    }
#pragma unroll
    for (int m = 16; m >= 1; m >>= 1) {
        ssq += __shfl_xor(ssq, m, 32);
        sd  += __shfl_xor(sd,  m, 32);
    }
    if (lane == 0) {
        const float INF = __builtin_inff();
        sq[row]    = ssq;
        dap[row]   = sd;
        minL[row]  = INF;
        fback[row] = INF;
    }
}

// ---------------------------------------------------------------------------
// Kernel 2: fused Gram-tile (fp32 WMMA 16x16x4) + distance + masked min mining.
// One wave per (16-row tile, 128-col strip). grid = (256, 32).
// ---------------------------------------------------------------------------
__global__ void __launch_bounds__(32) mine_kernel(
    const float* __restrict__ anchor, const int* __restrict__ labels,
    const float* __restrict__ sq, const float* __restrict__ dap,
    float* __restrict__ minL, float* __restrict__ fback)
{
    const float INF = __builtin_inff();
    const int lane = threadIdx.x & 31;
    const int half = lane >> 4;   // 0: K=0,1 / rows 0-7 of C ; 1: K=2,3 / rows 8-15
    const int lsub = lane & 15;
    const int i0 = blockIdx.x * 16;          // row-tile base
    const int j0 = blockIdx.y * (16 * NT);   // col-strip base

    // A fragment base: lane lsub holds row i0+lsub; half selects K pair (ISA A 16x4 layout)
    const float* aptr = anchor + (size_t)(i0 + lsub) * D_DIM + 2 * half;
    // B fragment base: lane lsub holds column j0+lsub; half selects K pair (B 4x16 layout)
    const float* bptr = anchor + (size_t)(j0 + lsub) * D_DIM + 2 * half;

    v8f acc[NT];
#pragma unroll
    for (int t = 0; t < NT; ++t) {
        v8f z = {0.f, 0.f, 0.f, 0.f, 0.f, 0.f, 0.f, 0.f};
        acc[t] = z;
    }

#pragma unroll 2
    for (int k = 0; k < D_DIM; k += 4) {
        v2f a = *(const v2f*)(aptr + k);
#pragma unroll
        for (int t = 0; t < NT; ++t) {
            v2f b = *(const v2f*)(bptr + (size_t)t * 16 * D_DIM + k);
            acc[t] = __builtin_amdgcn_wmma_f32_16x16x4_f32(
                /*neg_a=*/false, a, /*neg_b=*/false, b,
                /*c_mod=*/(short)0, acc[t], /*reuse_a=*/false, /*reuse_b=*/false);
        }
    }

    // Per-lane row-side data: C layout -> this lane's rows are i0 + v + 8*half
    float sqi[8], dapi[8];
    int   labi[8];
#pragma unroll
    for (int v = 0; v < 8; ++v) {
        int r   = i0 + v + 8 * half;
        sqi[v]  = sq[r];
        dapi[v] = dap[r];
        labi[v] = labels[r];
    }

    float mloc[8], floc[8];
#pragma unroll
    for (int v = 0; v < 8; ++v) { mloc[v] = INF; floc[v] = INF; }

#pragma unroll
    for (int t = 0; t < NT; ++t) {
        int   j  = j0 + t * 16 + lsub;   // C layout: N = lsub
        float sj = sq[j];
        int   lj = labels[j];
#pragma unroll
        for (int v = 0; v < 8; ++v) {
            float g  = acc[t][v];
            float d2 = fmaxf(sqi[v] + sj - 2.0f * g, 0.0f);
            float pd = sqrtf(d2);
            bool  diff = (lj != labi[v]);
            float cand = (diff && (pd > dapi[v])) ? pd : INF;
            float fb   = diff ? pd : INF;
            mloc[v] = fminf(mloc[v], cand);
            floc[v] = fminf(floc[v], fb);
        }
    }

    // Min-reduce across the 16 lanes of each half (xor masks stay within a half)
#pragma unroll
    for (int m = 8; m >= 1; m >>= 1) {
#pragma unroll
        for (int v = 0; v < 8; ++v) {
            mloc[v] = fminf(mloc[v], __shfl_xor(mloc[v], m, 32));
            floc[v] = fminf(floc[v], __shfl_xor(floc[v], m, 32));
        }
    }
    if (lsub == 0) {
#pragma unroll
        for (int v = 0; v < 8; ++v) {
            int r = i0 + v + 8 * half;
            atomic_min_f32_nonneg(&minL[r],  mloc[v]);
            atomic_min_f32_nonneg(&fback[r], floc[v]);
        }
    }
}

// ---------------------------------------------------------------------------
// Kernel 3: final per-row loss + mean reduction. Single block.
// ---------------------------------------------------------------------------
__global__ void __launch_bounds__(256) finalize_kernel(
    const float* __restrict__ dap, const float* __restrict__ minL,
    const float* __restrict__ fback, float* __restrict__ out)
{
    __shared__ float red[256];
    const float INF = __builtin_inff();
    int tid = threadIdx.x;
    float s = 0.0f;
    for (int r = tid; r < B_ROWS; r += 256) {
        float ml  = minL[r];
        float dan = (ml < INF) ? ml : fback[r];
        s += fmaxf(dap[r] - dan + MARGIN, 0.0f);
    }
    red[tid] = s;
    __syncthreads();
    for (int o = 128; o > 0; o >>= 1) {
        if (tid < o) red[tid] += red[tid + o];
        __syncthreads();
    }
    if (tid == 0) out[0] = red[0] / (float)B_ROWS;
}

extern "C" void kernel_launch(void* const* d_in, const int* in_sizes, int n_in,
                              void* d_out, int out_size, void* d_ws, size_t ws_size,
                              hipStream_t stream) {
    const float* anchor   = (const float*)d_in[0];
    const float* positive = (const float*)d_in[1];
    const int*   labels   = (const int*)d_in[2];
    float* out = (float*)d_out;

    float* ws    = (float*)d_ws;
    float* sq    = ws;
    float* dap   = ws + B_ROWS;
    float* minL  = ws + 2 * B_ROWS;
    float* fback = ws + 3 * B_ROWS;

    prep_kernel<<<B_ROWS / 8, 256, 0, stream>>>(anchor, positive, sq, dap, minL, fback);

    dim3 grid(B_ROWS / 16, B_ROWS / (16 * NT));
    mine_kernel<<<grid, 32, 0, stream>>>(anchor, labels, sq, dap, minL, fback);

    finalize_kernel<<<1, 256, 0, stream>>>(dap, minL, fback, out);
}